// OptimalTransportLayer_86002425135379
// MI455X (gfx1250) — compile-verified
//
#include <hip/hip_runtime.h>
#include <math.h>

// ---------------- Problem constants ----------------
#define B_        32
#define H_        512
#define W_        512
#define SPLITS    4
#define NBLK      (B_ * SPLITS)        // 128 persistent blocks
#define NTHR      512                  // 16 waves (wave32)
#define ROWS_PS   (H_ / SPLITS)        // 128 rows per block
#define COLS_PS   (W_ / SPLITS)        // 128 cols per block
#define MAXIT     100
#define EPS_      1e-5f
#define RVAL      (1.0f / (float)H_)
#define CVAL      (1.0f / (float)W_)

typedef __attribute__((ext_vector_type(2))) float v2f;
typedef __attribute__((ext_vector_type(8))) float v8f;

// ---------------- Global (grid-wide) sense-reversing barrier ----------------
__device__ __forceinline__ void global_barrier(unsigned* cnt, unsigned* gen) {
    __syncthreads();
    if (threadIdx.x == 0) {
        unsigned g = __hip_atomic_load(gen, __ATOMIC_ACQUIRE, __HIP_MEMORY_SCOPE_AGENT);
        unsigned a = __hip_atomic_fetch_add(cnt, 1u, __ATOMIC_ACQ_REL, __HIP_MEMORY_SCOPE_AGENT);
        if (a == NBLK - 1u) {
            __hip_atomic_store(cnt, 0u, __ATOMIC_RELAXED, __HIP_MEMORY_SCOPE_AGENT);
            __hip_atomic_fetch_add(gen, 1u, __ATOMIC_RELEASE, __HIP_MEMORY_SCOPE_AGENT);
        } else {
            while (__hip_atomic_load(gen, __ATOMIC_ACQUIRE, __HIP_MEMORY_SCOPE_AGENT) == g) {
                __builtin_amdgcn_s_sleep(1);
            }
        }
    }
    __syncthreads();
}

// ---------------- Init: zero control words, seed prev-buffers so u=v=1 ----------------
__global__ void sinkhorn_init(float* ws_s, float* ws_t, unsigned* ctl) {
    int tid = threadIdx.x;
    for (int i = tid; i < 2 + 2 * MAXIT + 8; i += blockDim.x) ctl[i] = 0u;
    float* s1 = ws_s + B_ * H_;   // buffer index 1 ("prev" for iteration 0)
    float* t1 = ws_t + B_ * W_;
    for (int i = tid; i < B_ * H_; i += blockDim.x) s1[i] = RVAL;  // r/s = 1
    for (int i = tid; i < B_ * W_; i += blockDim.x) t1[i] = CVAL;  // c/t = 1
}

// ---------------- Main persistent Sinkhorn kernel ----------------
__global__ __launch_bounds__(NTHR, 1)
void sinkhorn_main(const float* __restrict__ M, float* __restrict__ out,
                   float* __restrict__ ws_s, float* __restrict__ ws_t,
                   unsigned* __restrict__ ctl) {
    const int b    = blockIdx.x / SPLITS;
    const int sp   = blockIdx.x % SPLITS;
    const int tid  = threadIdx.x;
    const int lane = tid & 31;
    const int wave = tid >> 5;              // 0..15

    unsigned* bar_cnt = ctl;
    unsigned* bar_gen = ctl + 1;
    unsigned* viol    = ctl + 2;            // one flag per half-iteration, pre-zeroed

    const float* Mb = M   + (size_t)b * H_ * W_;
    float*       K  = out + (size_t)b * H_ * W_;   // K = exp(-M) lives in d_out

    __shared__ float sh_a[W_];   // phase A: v-vector | phase B: u'-vector | final: v
    __shared__ float sh_b[W_];   // phase B: t partials | final: u slice

    // ---- Phase 0: K = exp(-M) for this block's 128 rows (vectorized) ----
    {
        const int R0 = sp * ROWS_PS;
        const float4* m4 = (const float4*)(Mb + (size_t)R0 * W_);
        float4*       k4 = (float4*)(K + (size_t)R0 * W_);
        for (int i = tid; i < ROWS_PS * W_ / 4; i += NTHR) {
            float4 x = m4[i];
            float4 y;
            y.x = expf(-x.x); y.y = expf(-x.y); y.z = expf(-x.z); y.w = expf(-x.w);
            k4[i] = y;
        }
    }
    // No barrier needed: phase A of iter 0 only reads rows this block wrote.

    int exit_case = 0;          // 0 = maxiters, 1 = d1 break, 2 = d2 break
    int exit_it   = MAXIT;

    for (int it = 0; it < MAXIT; ++it) {
        const int cur = it & 1, prev = cur ^ 1;
        float*       s_cur  = ws_s + cur  * (B_ * H_) + b * H_;
        const float* s_prev = ws_s + prev * (B_ * H_) + b * H_;
        float*       t_cur  = ws_t + cur  * (B_ * W_) + b * W_;
        const float* t_prev = ws_t + prev * (B_ * W_) + b * W_;

        // ================= Phase A: s = K v,  v_j = c / t_prev[j] =================
        for (int j = tid; j < W_; j += NTHR) sh_a[j] = CVAL / t_prev[j];
        __syncthreads();
        {
            const int R0 = sp * ROWS_PS + wave * 8;
            unsigned lv = 0;
            for (int rr = 0; rr < 8; ++rr) {
                const int row = R0 + rr;
                const float* krow = K + (size_t)row * W_;
                float acc = 0.0f;
                #pragma unroll
                for (int k = 0; k < W_ / 32; ++k)
                    acc = fmaf(krow[lane + 32 * k], sh_a[lane + 32 * k], acc);
                #pragma unroll
                for (int off = 16; off > 0; off >>= 1) acc += __shfl_xor(acc, off, 32);
                if (lane == 0) {
                    s_cur[row] = acc;                       // commit s^(it)
                    float ucur = RVAL / s_prev[row];        // current row scaling
                    if (fabsf(ucur * acc - RVAL) > EPS_) lv = 1;
                }
            }
            if (lv) __hip_atomic_fetch_or(&viol[2 * it], 1u,
                                          __ATOMIC_RELAXED, __HIP_MEMORY_SCOPE_AGENT);
        }
        global_barrier(bar_cnt, bar_gen);
        if (__hip_atomic_load(&viol[2 * it], __ATOMIC_ACQUIRE,
                              __HIP_MEMORY_SCOPE_AGENT) == 0u) {
            exit_case = 1; exit_it = it; break;             // keep old u, old v
        }

        // ============ Phase B: t = K^T u',  u'_i = r / s_cur[i]  (WMMA) ============
        for (int i = tid; i < H_; i += NTHR) sh_a[i] = RVAL / s_cur[i];
        if (tid < COLS_PS) sh_b[tid] = 0.0f;
        __syncthreads();
        {
            const int g    = wave >> 1;                     // col-group 0..7 (16 cols each)
            const int wsub = wave & 1;                      // row-range half
            const int lcol = sp * COLS_PS + g * 16 + (lane & 15);
            const int hh   = (lane >> 4) << 1;              // 0 for lanes 0-15, 2 for 16-31
            v8f acc0 = {}; v8f acc1 = {};
            // Each wave: 256 rows of K, 64 WMMA f32 16x16x4 steps, 2-way interleaved.
            for (int k = 0; k < 64; k += 2) {
                {
                    const int rIdx = wsub * 256 + k * 4 + hh;
                    v2f a, bb;
                    a.x  = sh_a[rIdx];     a.y  = sh_a[rIdx + 1];      // u broadcast (all A rows equal)
                    bb.x = K[(size_t)rIdx * W_ + lcol];                 // B = row-major 4x16 tile of K
                    bb.y = K[(size_t)(rIdx + 1) * W_ + lcol];
                    acc0 = __builtin_amdgcn_wmma_f32_16x16x4_f32(
                        false, a, false, bb, (short)0, acc0, false, false);
                }
                {
                    const int rIdx = wsub * 256 + (k + 1) * 4 + hh;
                    v2f a, bb;
                    a.x  = sh_a[rIdx];     a.y  = sh_a[rIdx + 1];
                    bb.x = K[(size_t)rIdx * W_ + lcol];
                    bb.y = K[(size_t)(rIdx + 1) * W_ + lcol];
                    acc1 = __builtin_amdgcn_wmma_f32_16x16x4_f32(
                        false, a, false, bb, (short)0, acc1, false, false);
                }
            }
            // D rows are all identical; lanes 0-15 of VGPR0 hold t-partials for 16 cols.
            float tp = acc0[0] + acc1[0];
            if (lane < 16) atomicAdd(&sh_b[g * 16 + lane], tp);
        }
        __syncthreads();
        if (tid < COLS_PS) {
            const int jg = sp * COLS_PS + tid;
            float tj = sh_b[tid];
            t_cur[jg] = tj;                                  // commit t^(it)
            float vcur = CVAL / t_prev[jg];                  // current col scaling
            if (fabsf(vcur * tj - CVAL) > EPS_)
                __hip_atomic_fetch_or(&viol[2 * it + 1], 1u,
                                      __ATOMIC_RELAXED, __HIP_MEMORY_SCOPE_AGENT);
        }
        global_barrier(bar_cnt, bar_gen);
        if (__hip_atomic_load(&viol[2 * it + 1], __ATOMIC_ACQUIRE,
                              __HIP_MEMORY_SCOPE_AGENT) == 0u) {
            exit_case = 2; exit_it = it; break;              // keep new u, old v
        }
    }

    // ---- Select which s/t buffers define the final u, v ----
    int su, tv;
    if (exit_case == 1)      { su = (exit_it + 1) & 1; tv = (exit_it + 1) & 1; } // P  = u,v (old)
    else if (exit_case == 2) { su =  exit_it      & 1; tv = (exit_it + 1) & 1; } // Pr = u',v (old)
    else                     { su = (MAXIT - 1) & 1;   tv = (MAXIT - 1) & 1;   } // Pc after 100 iters

    // ---- Finalize in place: out[i][j] = u_i * K[i][j] * v_j (this block's rows) ----
    __syncthreads();
    {
        const float* sU = ws_s + su * (B_ * H_) + b * H_;
        const float* tV = ws_t + tv * (B_ * W_) + b * W_;
        for (int j = tid; j < W_; j += NTHR) sh_a[j] = CVAL / tV[j];
        if (tid < ROWS_PS) sh_b[tid] = RVAL / sU[sp * ROWS_PS + tid];
        __syncthreads();
        const int R0 = sp * ROWS_PS;
        for (int idx = tid; idx < ROWS_PS * W_; idx += NTHR) {
            const int r = idx >> 9;        // / 512 (W_)
            const int j = idx & (W_ - 1);
            const size_t off = (size_t)(R0 + r) * W_ + j;
            K[off] = sh_b[r] * K[off] * sh_a[j];
        }
    }
}

// ---------------- Host launcher ----------------
extern "C" void kernel_launch(void* const* d_in, const int* in_sizes, int n_in,
                              void* d_out, int out_size, void* d_ws, size_t ws_size,
                              hipStream_t stream) {
    const float* M  = (const float*)d_in[0];
    float* out      = (float*)d_out;
    float* ws_s     = (float*)d_ws;               // [2][B][H]
    float* ws_t     = ws_s + 2 * B_ * H_;         // [2][B][W]
    unsigned* ctl   = (unsigned*)(ws_t + 2 * B_ * W_);  // cnt, gen, viol[2*MAXIT]

    sinkhorn_init<<<dim3(1), dim3(1024), 0, stream>>>(ws_s, ws_t, ctl);
    sinkhorn_main<<<dim3(NBLK), dim3(NTHR), 0, stream>>>(M, out, ws_s, ws_t, ctl);
}